// QuantumFeedForward_45981919871259
// MI455X (gfx1250) — compile-verified
//
#include <hip/hip_runtime.h>
#include <math.h>

typedef float v2f __attribute__((ext_vector_type(2)));
typedef float v8f __attribute__((ext_vector_type(8)));

#define D_DIM 512
#define H_DIM 2048

// Pack the 4 live columns of W1 (D x H, row-major) into contiguous w1c[D][4].
__global__ __launch_bounds__(256)
void qff_pack_w1(const float* __restrict__ W1, float* __restrict__ w1c) {
  int idx = blockIdx.x * blockDim.x + threadIdx.x;
  if (idx < D_DIM * 4) {
    int d = idx >> 2, j = idx & 3;
    w1c[idx] = W1[d * H_DIM + j];
  }
}

// One wave (32 lanes) handles a tile of 16 rows end-to-end.
__global__ __launch_bounds__(32)
void qff_main(const float* __restrict__ x, const float* __restrict__ w1c,
              const float* __restrict__ b1, const float* __restrict__ qw,
              const float* __restrict__ Wp, const float* __restrict__ bp,
              float* __restrict__ out)
{
  __shared__ float sAng[64];  // angles[16 rows][4 qubits]
  __shared__ float sQ[64];    // expvals[16 rows][4 qubits]
  __shared__ float sG[64];    // 8 Rot gates x (m00,m01,m10,m11) complex

  const int lane = threadIdx.x;
  const int hsel = lane >> 4;   // 0: lanes 0-15, 1: lanes 16-31
  const int lid  = lane & 15;
  const long row0 = (long)blockIdx.x * 16;

  // ---- Rot gate matrices: uniform over all rows, computed once per wave ----
  if (lane < 8) {
    const int l = lane >> 2, i = lane & 3;
    const float phi = qw[(l * 4 + i) * 3 + 0];
    const float th  = qw[(l * 4 + i) * 3 + 1];
    const float om  = qw[(l * 4 + i) * 3 + 2];
    const float c = cosf(0.5f * th), s = sinf(0.5f * th);
    const float a = -0.5f * (phi + om);   // ep = exp(i*a)
    const float b =  0.5f * (phi - om);   // em = exp(i*b)
    const float ca = cosf(a), sa = sinf(a);
    const float cb = cosf(b), sb = sinf(b);
    float* g = &sG[lane * 8];
    g[0] =  c * ca;  g[1] =  c * sa;   // m00 = ep*c
    g[2] = -s * cb;  g[3] = -s * sb;   // m01 = -em*s
    g[4] =  s * cb;  g[5] = -s * sb;   // m10 = conj(em)*s
    g[6] =  c * ca;  g[7] = -c * sa;   // m11 = conj(ep)*c
  }

  // ---- GEMM1: H[16x16] = Xtile[16x512] * W1c[512x(4 pad to 16)]  (128 WMMAs) ----
  // A layout (16x4 f32): lane<16 -> row=lid, K=0,1 ; lane>=16 -> row=lid, K=2,3
  // B layout (4x16 f32): lane<16 -> col=lid, K=0,1 ; lane>=16 -> col=lid, K=2,3
  v8f acc = {};
  const float* xrow = x + (row0 + lid) * D_DIM + 2 * hsel;
  const float bsel = (lid < 4) ? 1.0f : 0.0f;   // zero-pad columns 4..15
  const int   bcol = lid & 3;
  #pragma unroll 4
  for (int k = 0; k < D_DIM; k += 4) {
    v2f afrag; afrag.x = xrow[k]; afrag.y = xrow[k + 1];
    v2f bfrag;
    bfrag.x = bsel * w1c[(k + 2 * hsel + 0) * 4 + bcol];
    bfrag.y = bsel * w1c[(k + 2 * hsel + 1) * 4 + bcol];
    acc = __builtin_amdgcn_wmma_f32_16x16x4_f32(false, afrag, false, bfrag,
                                                (short)0, acc, false, false);
  }

  // ---- bias + exact GeLU on the 4 live columns -> angles in LDS ----
  // C/D layout: lane holds col n=lid, rows m = hsel*8 + r
  if (lid < 4) {
    const float bias = b1[lid];
    #pragma unroll
    for (int r = 0; r < 8; ++r) {
      const float v = acc[r] + bias;
      sAng[(hsel * 8 + r) * 4 + lid] = 0.5f * v * (1.0f + erff(v * 0.70710678118654752f));
    }
  }
  __syncthreads();

  // ---- 4-qubit statevector sim: basis state b = lid, two rows per iteration ----
  for (int t = 0; t < 8; ++t) {
    const int row = 2 * t + hsel;
    const int basis = lid;
    // initial product state amplitudes (real)
    float re = 1.0f, im = 0.0f;
    #pragma unroll
    for (int i = 0; i < 4; ++i) {
      const float a2 = 0.5f * sAng[row * 4 + i];
      re *= ((basis >> (3 - i)) & 1) ? sinf(a2) : cosf(a2);
    }
    #pragma unroll
    for (int l = 0; l < 2; ++l) {
      // single-qubit Rot on each qubit
      #pragma unroll
      for (int i = 0; i < 4; ++i) {
        const float* g = &sG[(l * 4 + i) * 8];
        const int mask = 8 >> i;                   // qubit 0 = MSB of 4-bit basis
        const int tb = (basis >> (3 - i)) & 1;
        const float pr = __shfl_xor(re, mask, 32);
        const float pi = __shfl_xor(im, mask, 32);
        const float nr0 = g[0]*re - g[1]*im + g[2]*pr - g[3]*pi;  // bit=0: m00*a0+m01*a1
        const float ni0 = g[0]*im + g[1]*re + g[2]*pi + g[3]*pr;
        const float nr1 = g[4]*pr - g[5]*pi + g[6]*re - g[7]*im;  // bit=1: m10*a0+m11*a1
        const float ni1 = g[4]*pi + g[5]*pr + g[6]*im + g[7]*re;
        re = tb ? nr1 : nr0;
        im = tb ? ni1 : ni0;
      }
      // ring entangler: CNOT(i, (i + r) % 4), r = (l % 3) + 1
      const int rsh = (l % 3) + 1;
      #pragma unroll
      for (int i = 0; i < 4; ++i) {
        const int tq = (i + rsh) & 3;
        const int maskT = 8 >> tq;
        const int ctrl = (basis >> (3 - i)) & 1;
        const float pr = __shfl_xor(re, maskT, 32);
        const float pi = __shfl_xor(im, maskT, 32);
        re = ctrl ? pr : re;
        im = ctrl ? pi : im;
      }
    }
    // PauliZ expectation values: signed butterfly reductions over 16 lanes
    const float prob = re * re + im * im;
    #pragma unroll
    for (int i = 0; i < 4; ++i) {
      float sv = ((basis >> (3 - i)) & 1) ? -prob : prob;
      sv += __shfl_xor(sv, 1, 32);
      sv += __shfl_xor(sv, 2, 32);
      sv += __shfl_xor(sv, 4, 32);
      sv += __shfl_xor(sv, 8, 32);
      if (lid == 0) sQ[row * 4 + i] = sv;
    }
  }
  __syncthreads();

  // ---- GEMM2: Out[16x512] = Q[16x4] * Wp[4x512] + bp   (32 WMMAs) ----
  v2f aq;
  aq.x = sQ[lid * 4 + 2 * hsel + 0];
  aq.y = sQ[lid * 4 + 2 * hsel + 1];
  for (int n0 = 0; n0 < D_DIM; n0 += 16) {
    v2f bw;
    bw.x = Wp[(2 * hsel + 0) * D_DIM + n0 + lid];
    bw.y = Wp[(2 * hsel + 1) * D_DIM + n0 + lid];
    v8f c = {};
    c = __builtin_amdgcn_wmma_f32_16x16x4_f32(false, aq, false, bw,
                                              (short)0, c, false, false);
    const float bias = bp[n0 + lid];
    float* orow = out + (row0 + hsel * 8) * D_DIM + n0 + lid;
    #pragma unroll
    for (int r = 0; r < 8; ++r)
      orow[(long)r * D_DIM] = c[r] + bias;
  }
}

extern "C" void kernel_launch(void* const* d_in, const int* in_sizes, int n_in,
                              void* d_out, int out_size, void* d_ws, size_t ws_size,
                              hipStream_t stream) {
  const float* x  = (const float*)d_in[0];   // [B,S,512]
  const float* W1 = (const float*)d_in[1];   // [512,2048]
  const float* b1 = (const float*)d_in[2];   // [2048]
  const float* qw = (const float*)d_in[3];   // [2,4,3]
  const float* Wp = (const float*)d_in[4];   // [4,512]
  const float* bp = (const float*)d_in[5];   // [512]
  float* out = (float*)d_out;                // [B,S,512]
  float* w1c = (float*)d_ws;                 // 2048 floats (8 KB) scratch

  const int rows = in_sizes[0] / D_DIM;      // B*S = 32768 (multiple of 16)
  qff_pack_w1<<<(D_DIM * 4 + 255) / 256, 256, 0, stream>>>(W1, w1c);
  qff_main<<<rows / 16, 32, 0, stream>>>(x, w1c, b1, qw, Wp, bp, out);
}